// BaseMergedLinear_58076547776655
// MI455X (gfx1250) — compile-verified
//
#include <hip/hip_runtime.h>
#include <hip/hip_bf16.h>

// LoRA merged-linear:  y = x @ (scatter(B_g A_g))^T, computed low-rank:
//   T = x @ A^T  (1024 x 32)   then   y[:,block_g] = T_g @ B_g^T,  y[:,block1]=0
// fp32 throughout (matches reference); roofline: ~67 MB HBM traffic, dominated
// by the 50 MB output write.  Stage 1 is K-split 16 ways for wave parallelism.

typedef float v2f __attribute__((ext_vector_type(2)));
typedef float v4f __attribute__((ext_vector_type(4)));
typedef float v8f __attribute__((ext_vector_type(8)));

#define IN_F   4096
#define OUT_F  12288
#define OUT_PG 4096
#define RNK    16
#define MTOT   1024          // 2 * 512 flattened rows of x
#define KSEG   16            // K-split factor for stage 1
#define KCH    (IN_F / KSEG) // 256 K-columns per segment
#define TSZ    (MTOT * 32)   // elements in one T (or T-partial) buffer

// ---------------------------------------------------------------------------
// Stage 1: Tpart[seg][m][g*16+n] = sum_{k in seg} x[m][k] * wA[g*16+n][k]
// One wave = one (16-row M-tile, K-segment) pair -> 64 x 16 = 1024 waves.
// WMMA f32 16x16x4:  A operand (x tile, 16x4): lane = M row, lane-half -> K{0,2},
// vgpr1 -> K+1.  B operand (A_g^T, 4x16): lane = N col, same K pairing.
// ---------------------------------------------------------------------------
__global__ __launch_bounds__(128) void lora_stage1(const float* __restrict__ x,
                                                   const float* __restrict__ wA,
                                                   float* __restrict__ Tpart) {
  const int lane    = threadIdx.x & 31;
  const int wave    = threadIdx.x >> 5;
  const int mtile   = blockIdx.x * 4 + wave;       // 0..63
  const int seg     = blockIdx.y;                  // 0..KSEG-1
  const int kBase   = seg * KCH;
  const int rowBase = mtile * 16;
  const int laneLo  = lane & 15;
  const int laneHiK = (lane >> 4) * 2;             // 0 or 2

  const float* xrow  = x  + (size_t)(rowBase + laneLo) * IN_F + kBase + laneHiK;
  const float* a0row = wA + (size_t)(laneLo)           * IN_F + kBase + laneHiK; // group 0
  const float* a1row = wA + (size_t)(16 + laneLo)      * IN_F + kBase + laneHiK; // group 1

  v8f acc0 = {};
  v8f acc1 = {};
  #pragma unroll 4
  for (int k = 0; k < KCH; k += 4) {
    v2f a  = *(const v2f*)(xrow  + k);
    v2f b0 = *(const v2f*)(a0row + k);
    v2f b1 = *(const v2f*)(a1row + k);
    acc0 = __builtin_amdgcn_wmma_f32_16x16x4_f32(false, a, false, b0, (short)0, acc0, false, false);
    acc1 = __builtin_amdgcn_wmma_f32_16x16x4_f32(false, a, false, b1, (short)0, acc1, false, false);
  }

  // C/D layout: vgpr r, lanes 0-15 -> M=r, lanes 16-31 -> M=r+8; N = lane&15.
  const int mOff = (lane >> 4) * 8;
  float* t0 = Tpart + (size_t)seg * TSZ + (size_t)(rowBase + mOff) * 32 + laneLo;
  #pragma unroll
  for (int r = 0; r < 8; ++r) {
    t0[(size_t)r * 32]      = acc0[r];
    t0[(size_t)r * 32 + 16] = acc1[r];
  }
}

// ---------------------------------------------------------------------------
// Reduce the KSEG partial T buffers (fixed order -> bitwise deterministic).
// ---------------------------------------------------------------------------
__global__ __launch_bounds__(256) void lora_reduce(const float* __restrict__ Tpart,
                                                   float* __restrict__ T) {
  const int i = blockIdx.x * 256 + threadIdx.x;    // 0..TSZ-1
  float s = 0.0f;
  #pragma unroll
  for (int seg = 0; seg < KSEG; ++seg)
    s += Tpart[(size_t)seg * TSZ + i];
  T[i] = s;
}

// ---------------------------------------------------------------------------
// Stage 2: y tile = T_g tile (16 x 16) @ wB_g^T slice (16 x 16), or zeros for
// the disabled K block.  One wave per 16x16 output tile; 8 waves per block
// cover 128 output columns.  Accumulator is transposed through LDS so stores
// go out as global_store_b128 (output write bandwidth is the bottleneck).
// ---------------------------------------------------------------------------
#define LDS_STRIDE 20   // floats per row: 80 B, keeps b128 rows 16B-aligned

__global__ __launch_bounds__(256) void lora_stage2(const float* __restrict__ T,
                                                   const float* __restrict__ wB,
                                                   float* __restrict__ out) {
  __shared__ float lds[8 * 16 * LDS_STRIDE];

  const int lane    = threadIdx.x & 31;
  const int wave    = threadIdx.x >> 5;
  const int laneLo  = lane & 15;
  const int laneHiK = (lane >> 4) * 2;
  const int blk     = blockIdx.z;                      // 0=Q, 1=K(zeros), 2=V
  const int rowBase = blockIdx.y * 16;
  const int colBase = (blockIdx.x * 8 + wave) * 16;    // within the 4096-col block

  float* outTile = out + (size_t)rowBase * OUT_F + (size_t)blk * OUT_PG + colBase;

  const int c0 = lane;        // chunk ids: row = c>>2, 4-float segment = c&3
  const int c1 = lane + 32;

  if (blk == 1) {             // disabled block: stream zeros, b128 per lane
    v4f z = {};
    *(v4f*)(outTile + (size_t)(c0 >> 2) * OUT_F + (c0 & 3) * 4) = z;
    *(v4f*)(outTile + (size_t)(c1 >> 2) * OUT_F + (c1 & 3) * 4) = z;
    return;
  }
  const int g = (blk == 0) ? 0 : 1;

  // A operand: T rows rowBase..+15, K = g*16 .. g*16+15
  const float* tRow = T  + (size_t)(rowBase + laneLo) * 32 + g * RNK + laneHiK;
  // B operand: B^T[k][n] = wB[g*4096 + colBase + n][k]
  const float* bRow = wB + (size_t)(g * OUT_PG + colBase + laneLo) * RNK + laneHiK;

  v8f acc = {};
  #pragma unroll
  for (int k4 = 0; k4 < 4; ++k4) {
    v2f a = *(const v2f*)(tRow + k4 * 4);
    v2f b = *(const v2f*)(bRow + k4 * 4);
    acc = __builtin_amdgcn_wmma_f32_16x16x4_f32(false, a, false, b, (short)0, acc, false, false);
  }

  // Accumulator -> LDS (row-major tile), then b128 row-major stores.
  float* myLds = lds + wave * (16 * LDS_STRIDE);
  const int mOff = (lane >> 4) * 8;
  #pragma unroll
  for (int r = 0; r < 8; ++r)
    myLds[(mOff + r) * LDS_STRIDE + laneLo] = acc[r];

  v4f d0 = *(const v4f*)(myLds + (c0 >> 2) * LDS_STRIDE + (c0 & 3) * 4);
  v4f d1 = *(const v4f*)(myLds + (c1 >> 2) * LDS_STRIDE + (c1 & 3) * 4);
  *(v4f*)(outTile + (size_t)(c0 >> 2) * OUT_F + (c0 & 3) * 4) = d0;
  *(v4f*)(outTile + (size_t)(c1 >> 2) * OUT_F + (c1 & 3) * 4) = d1;
}

extern "C" void kernel_launch(void* const* d_in, const int* in_sizes, int n_in,
                              void* d_out, int out_size, void* d_ws, size_t ws_size,
                              hipStream_t stream) {
  (void)in_sizes; (void)n_in; (void)out_size; (void)ws_size;
  const float* x  = (const float*)d_in[0];   // (2,512,4096) -> (1024, 4096)
  const float* wA = (const float*)d_in[1];   // (32, 4096)
  const float* wB = (const float*)d_in[2];   // (8192, 16)
  float* out = (float*)d_out;                // (1024, 12288)

  float* Tpart = (float*)d_ws;               // KSEG x (1024 x 32) partials, 2 MB
  float* T     = Tpart + (size_t)KSEG * TSZ; // reduced (1024 x 32), 128 KB

  // Stage 1: 64 M-tiles x 16 K-segments, 4 waves per block -> 1024 waves.
  lora_stage1<<<dim3(16, KSEG), dim3(128), 0, stream>>>(x, wA, Tpart);
  // Deterministic fixed-order reduction of partials.
  lora_reduce<<<dim3(TSZ / 256), dim3(256), 0, stream>>>(Tpart, T);
  // Stage 2: cols: 32 groups of 128; rows: 64 tiles; z: 3 output blocks.
  lora_stage2<<<dim3(32, 64, 3), dim3(256), 0, stream>>>(T, wB, out);
}